// TransitionUp_37091337568771
// MI455X (gfx1250) — compile-verified
//
#include <hip/hip_runtime.h>
#include <math.h>

typedef __attribute__((ext_vector_type(2))) float v2f;
typedef __attribute__((ext_vector_type(8))) float v8f;

namespace {
constexpr int BB   = 4;
constexpr int NP   = 8192;
constexpr int SP   = 2048;
constexpr int CSK  = 256;          // skip feature dim
constexpr int CLO  = 512;          // low-res feature dim
constexpr int CIN  = CSK + CLO;    // 768
constexpr int COUT = 256;
constexpr int MT   = 32;           // hi points per workgroup
constexpr int LDA  = CIN + 4;      // 772, % 64 == 4 -> conflict-free v2f A loads
constexpr int LDH  = COUT + 4;     // 260, % 64 == 4
constexpr int FUSED_F = MT * LDA;  // 24704 floats (region A)
constexpr int REGB_F  = MT * LDH;  // 8320 floats  (region B; >= 7680 KNN scratch)
constexpr int REGC_F  = 96 + 96 + 512 + 64; // idx, w, LN partials, LN stats
constexpr int SMEM_F  = FUSED_F + REGB_F + REGC_F;  // 33792 floats = 135168 B
constexpr float EPS_IDW = 1e-8f;
constexpr float EPS_LN  = 1e-5f;
// packed B-operand weights in d_ws
constexpr int PK_W1_F = CIN * COUT;    // 196608 floats
constexpr int PK_W2_F = COUT * COUT;   //  65536 floats
constexpr size_t PK_BYTES = (size_t)(PK_W1_F + PK_W2_F) * sizeof(float); // 1 MB
}

// ---------------------------------------------------------------------------
// Repack W1/W2 into WMMA B-operand tile order:
//   pk[kb4][cg][lane] = { W[kb4*4 + (lane>>4)*2][cg*16 + lane&15],
//                         W[kb4*4 + (lane>>4)*2 + 1][cg*16 + lane&15] }
// so the GEMM loop fetches one coalesced b64 per lane per 16x16x4 tile.
// ---------------------------------------------------------------------------
__global__ __launch_bounds__(256)
void pack_weights_kernel(const float* __restrict__ W1,
                         const float* __restrict__ W2,
                         float* __restrict__ pk)
{
    const int fi  = blockIdx.x * 256 + threadIdx.x;  // float2 index
    const int NW1 = PK_W1_F / 2;
    const int NW2 = PK_W2_F / 2;
    if (fi >= NW1 + NW2) return;
    const float* W;
    float* dst;
    int li;
    if (fi < NW1) { W = W1; dst = pk;           li = fi; }
    else          { W = W2; dst = pk + PK_W1_F; li = fi - NW1; }
    const int lane = li & 31;
    const int cg   = (li >> 5) & 15;
    const int kb4  = li >> 9;
    const int krow = kb4 * 4 + (lane >> 4) * 2;
    const int col  = cg * 16 + (lane & 15);
    v2f v;
    v.x = W[(size_t)krow * COUT + col];
    v.y = W[(size_t)(krow + 1) * COUT + col];
    *(v2f*)&dst[(size_t)li * 2] = v;
}

// ---------------------------------------------------------------------------
// Fused: KNN(K=3) -> IDW interp -> [skip|interp]@W1+b1,relu -> @W2+b2 -> LN
// ---------------------------------------------------------------------------
template <bool PACKED>
__global__ __launch_bounds__(256, 1)
void tu_fused_kernel(const float* __restrict__ xyz_hi,
                     const float* __restrict__ xyz_lo,
                     const float* __restrict__ feat_skip,
                     const float* __restrict__ feat_lo,
                     const float* __restrict__ W1,
                     const float* __restrict__ b1,
                     const float* __restrict__ W2,
                     const float* __restrict__ b2,
                     const float* __restrict__ gamma,
                     const float* __restrict__ beta,
                     const float* __restrict__ pW1,
                     const float* __restrict__ pW2,
                     float* __restrict__ out)
{
    extern __shared__ float smem[];
    float* fusedT = smem;                        // [MT][LDA]  region A
    float* regB   = smem + FUSED_F;              // region B
    float* xyzS   = regB;                        // [SP*3]           (KNN phase)
    float* pd     = regB + SP * 3;               // [MT*8*3]         (KNN phase)
    int*   piv    = (int*)(pd + MT * 8 * 3);     // [MT*8*3]         (KNN phase)
    float* hT     = regB;                        // [MT][LDH]  alias (GEMM phase)
    float* h2     = fusedT;                      // [MT][LDH]  alias (post-GEMM1)
    float* regC   = smem + FUSED_F + REGB_F;
    int*   kIdx   = (int*)regC;                  // [MT*3]
    float* kW     = regC + 96;                   // [MT*3]
    float* lnPart = regC + 192;                  // [MT*8*2]
    float* lnStat = regC + 704;                  // [MT*2]

    const int tid  = threadIdx.x;
    const int lane = tid & 31;
    const int wv   = tid >> 5;

    const int tilesPerB = NP / MT;
    const int b  = blockIdx.x / tilesPerB;
    const int m0 = (blockIdx.x % tilesPerB) * MT;

    // ---------------- stage xyz_lo in LDS ----------------
    {
        const float* src = xyz_lo + (size_t)b * SP * 3;
        for (int i = tid; i < SP * 3; i += 256) xyzS[i] = src[i];
    }
    __syncthreads();

    // ---------------- KNN (K=3), 8 threads per point ----------------
    const int p  = tid >> 3;   // point in tile
    const int sl = tid & 7;    // slice
    {
        const float* ph = xyz_hi + ((size_t)b * NP + m0 + p) * 3;
        const float hx = ph[0], hy = ph[1], hz = ph[2];
        float d0 = 3e38f, d1 = 3e38f, d2v = 3e38f;
        int   i0 = 0,    i1 = 0,    i2 = 0;
        for (int j = sl; j < SP; j += 8) {   // stride-8 -> conflict-free LDS reads
            const float dx = hx - xyzS[j*3+0];
            const float dy = hy - xyzS[j*3+1];
            const float dz = hz - xyzS[j*3+2];
            const float dd = dx*dx + dy*dy + dz*dz;
            if (dd < d2v) {
                if (dd < d0)      { d2v=d1; i2=i1; d1=d0; i1=i0; d0=dd; i0=j; }
                else if (dd < d1) { d2v=d1; i2=i1; d1=dd; i1=j; }
                else              { d2v=dd; i2=j; }
            }
        }
        const int base = (p*8 + sl)*3;
        pd[base+0]=d0;  pd[base+1]=d1;  pd[base+2]=d2v;
        piv[base+0]=i0; piv[base+1]=i1; piv[base+2]=i2;
    }
    __syncthreads();

    if (tid < MT) {   // merge 8 partial top-3 lists -> IDW weights
        float d0 = 3e38f, d1 = 3e38f, d2v = 3e38f;
        int   i0 = 0,    i1 = 0,    i2 = 0;
        for (int q = 0; q < 24; ++q) {
            const float dd = pd[tid*24 + q];
            const int   jj = piv[tid*24 + q];
            if (dd < d2v) {
                if (dd < d0)      { d2v=d1; i2=i1; d1=d0; i1=i0; d0=dd; i0=jj; }
                else if (dd < d1) { d2v=d1; i2=i1; d1=dd; i1=jj; }
                else              { d2v=dd; i2=jj; }
            }
        }
        const float w0 = 1.0f / (sqrtf(d0)  + EPS_IDW);
        const float w1v= 1.0f / (sqrtf(d1)  + EPS_IDW);
        const float w2v= 1.0f / (sqrtf(d2v) + EPS_IDW);
        const float inv = 1.0f / (w0 + w1v + w2v);
        kIdx[tid*3+0]=i0; kIdx[tid*3+1]=i1; kIdx[tid*3+2]=i2;
        kW[tid*3+0]=w0*inv; kW[tid*3+1]=w1v*inv; kW[tid*3+2]=w2v*inv;
    }
    __syncthreads();

    // ---------------- fused = [feat_skip | interp] tile in LDS ----------------
    {
        const float* fs = feat_skip + ((size_t)b * NP + m0 + p) * CSK;
        for (int c = sl*32; c < sl*32 + 32; c += 4)
            *(float4*)&fusedT[p*LDA + c] = *(const float4*)&fs[c];

        const float* fl = feat_lo + (size_t)b * SP * CLO;
        const float* r0 = fl + (size_t)kIdx[p*3+0] * CLO;
        const float* r1 = fl + (size_t)kIdx[p*3+1] * CLO;
        const float* r2 = fl + (size_t)kIdx[p*3+2] * CLO;
        const float w0 = kW[p*3+0], w1v = kW[p*3+1], w2v = kW[p*3+2];
        for (int c = sl*64; c < sl*64 + 64; c += 4) {
            const float4 a = *(const float4*)&r0[c];
            const float4 u = *(const float4*)&r1[c];
            const float4 v = *(const float4*)&r2[c];
            float4 o;
            o.x = w0*a.x + w1v*u.x + w2v*v.x;
            o.y = w0*a.y + w1v*u.y + w2v*v.y;
            o.z = w0*a.z + w1v*u.z + w2v*v.z;
            o.w = w0*a.w + w1v*u.w + w2v*v.w;
            *(float4*)&fusedT[p*LDA + CSK + c] = o;
        }
    }
    __syncthreads();

    // WMMA operand coordinates (16x16x4 f32: A lanes 0-15 rows, lanes 16-31 K+2)
    const int lm  = lane & 15;
    const int lkh = (lane >> 4) * 2;     // 0 or 2
    const int mb  = (wv & 1) * 16;       // row group
    const int cgb = (wv >> 1) * 4;       // 4 col tiles of 16

    // ---------------- GEMM1: h = relu(fused @ W1 + b1) ----------------
    {
        v8f acc[4] = {};
        const float* aRow = &fusedT[(mb + lm) * LDA];
        const float* bp   = PACKED ? (pW1 + (size_t)cgb * 64 + lane * 2) : nullptr;
        #pragma unroll 4
        for (int kb = 0; kb < CIN; kb += 4) {
            const int krow = kb + lkh;
            const v2f a = *(const v2f*)&aRow[krow];
            #pragma unroll
            for (int c = 0; c < 4; ++c) {
                v2f bv;
                if (PACKED) {
                    bv = *(const v2f*)(bp + c * 64);
                } else {
                    const int n = (cgb + c)*16 + lm;
                    bv.x = W1[(size_t)krow*COUT + n];
                    bv.y = W1[(size_t)(krow+1)*COUT + n];
                }
                acc[c] = __builtin_amdgcn_wmma_f32_16x16x4_f32(
                            false, a, false, bv, (short)0, acc[c], false, false);
            }
            if (PACKED) bp += 1024;   // next k-step panel (16 tiles * 64 floats)
        }
        #pragma unroll
        for (int c = 0; c < 4; ++c) {
            const int n = (cgb + c)*16 + lm;
            const float bia = b1[n];
            #pragma unroll
            for (int r = 0; r < 8; ++r) {
                float v = acc[c][r] + bia;
                hT[(mb + (lane >> 4)*8 + r)*LDH + n] = v > 0.f ? v : 0.f;
            }
        }
    }
    __syncthreads();

    // ---------------- GEMM2: h2 = h @ W2 + b2 ----------------
    {
        v8f acc[4] = {};
        const float* aRow = &hT[(mb + lm) * LDH];
        const float* bp   = PACKED ? (pW2 + (size_t)cgb * 64 + lane * 2) : nullptr;
        #pragma unroll 4
        for (int kb = 0; kb < COUT; kb += 4) {
            const int krow = kb + lkh;
            const v2f a = *(const v2f*)&aRow[krow];
            #pragma unroll
            for (int c = 0; c < 4; ++c) {
                v2f bv;
                if (PACKED) {
                    bv = *(const v2f*)(bp + c * 64);
                } else {
                    const int n = (cgb + c)*16 + lm;
                    bv.x = W2[(size_t)krow*COUT + n];
                    bv.y = W2[(size_t)(krow+1)*COUT + n];
                }
                acc[c] = __builtin_amdgcn_wmma_f32_16x16x4_f32(
                            false, a, false, bv, (short)0, acc[c], false, false);
            }
            if (PACKED) bp += 1024;
        }
        #pragma unroll
        for (int c = 0; c < 4; ++c) {
            const int n = (cgb + c)*16 + lm;
            const float bia = b2[n];
            #pragma unroll
            for (int r = 0; r < 8; ++r)
                h2[(mb + (lane >> 4)*8 + r)*LDH + n] = acc[c][r] + bia;
        }
    }
    __syncthreads();

    // ---------------- LayerNorm + store ----------------
    {
        float s = 0.f, q = 0.f;
        for (int c = sl*32; c < sl*32 + 32; ++c) {
            const float v = h2[p*LDH + c];
            s += v; q += v*v;
        }
        lnPart[(p*8+sl)*2+0] = s;
        lnPart[(p*8+sl)*2+1] = q;
    }
    __syncthreads();
    if (tid < MT) {
        float s = 0.f, q = 0.f;
        for (int e = 0; e < 8; ++e) {
            s += lnPart[(tid*8+e)*2+0];
            q += lnPart[(tid*8+e)*2+1];
        }
        const float mu = s * (1.0f / COUT);
        float var = q * (1.0f / COUT) - mu*mu;
        var = var > 0.f ? var : 0.f;
        lnStat[tid*2+0] = mu;
        lnStat[tid*2+1] = rsqrtf(var + EPS_LN);
    }
    __syncthreads();
    {
        const float mu = lnStat[p*2+0], rs = lnStat[p*2+1];
        float* op = out + ((size_t)b*NP + m0 + p) * COUT;
        for (int c = sl*32; c < sl*32 + 32; c += 4) {
            const float4 v  = *(const float4*)&h2[p*LDH + c];
            const float4 g  = *(const float4*)&gamma[c];
            const float4 be = *(const float4*)&beta[c];
            float4 o;
            o.x = (v.x - mu)*rs*g.x + be.x;
            o.y = (v.y - mu)*rs*g.y + be.y;
            o.z = (v.z - mu)*rs*g.z + be.z;
            o.w = (v.w - mu)*rs*g.w + be.w;
            *(float4*)&op[c] = o;
        }
    }
}

extern "C" void kernel_launch(void* const* d_in, const int* in_sizes, int n_in,
                              void* d_out, int out_size, void* d_ws, size_t ws_size,
                              hipStream_t stream)
{
    (void)in_sizes; (void)n_in; (void)out_size;
    const float* xyz_hi    = (const float*)d_in[0];
    const float* xyz_lo    = (const float*)d_in[1];
    const float* feat_skip = (const float*)d_in[2];
    const float* feat_lo   = (const float*)d_in[3];
    const float* W1        = (const float*)d_in[4];
    const float* b1        = (const float*)d_in[5];
    const float* W2        = (const float*)d_in[6];
    const float* b2        = (const float*)d_in[7];
    const float* gamma     = (const float*)d_in[8];
    const float* beta      = (const float*)d_in[9];
    float* out = (float*)d_out;

    const dim3 grid(BB * NP / MT);   // 1024 workgroups
    const dim3 block(256);           // 8 waves (wave32)
    const size_t shmem = (size_t)SMEM_F * sizeof(float);

    const bool usePacked = (d_ws != nullptr) && (ws_size >= PK_BYTES);
    if (usePacked) {
        float* pk = (float*)d_ws;
        const int n2 = (PK_W1_F + PK_W2_F) / 2;              // float2 elements
        pack_weights_kernel<<<dim3((n2 + 255) / 256), dim3(256), 0, stream>>>(W1, W2, pk);
        tu_fused_kernel<true><<<grid, block, shmem, stream>>>(
            xyz_hi, xyz_lo, feat_skip, feat_lo, W1, b1, W2, b2, gamma, beta,
            pk, pk + PK_W1_F, out);
    } else {
        tu_fused_kernel<false><<<grid, block, shmem, stream>>>(
            xyz_hi, xyz_lo, feat_skip, feat_lo, W1, b1, W2, b2, gamma, beta,
            nullptr, nullptr, out);
    }
}